// Euler_Maruyama_30133490548782
// MI455X (gfx1250) — compile-verified
//
#include <hip/hip_runtime.h>
#include <cstdint>
#include <cstddef>

// ---------------------------------------------------------------------------
// Euler–Maruyama prefix-product scan, MI455X (gfx1250).
//
//   z[i+1] = z[i] * (c0 + c1 * w[i+1]),  c0 = 1 + r*dt, c1 = s*sqrt(dt)
//
// Memory-bound (~0.4 FLOP/byte).  gfx1250 plan:
//   * 1 thread per path; each wave owns 32 consecutive rows (lane == row).
//   * W/Z rows have stride 4100 B (1025 f32, payload at col 1) -> per-step
//     accesses are 4 B scatters.  Fix with LDS tiles of 32 rows x TK=32 steps:
//       - LOAD: one Tensor-Data-Mover descriptor per tile per wave
//         (tensor_load_to_lds): 2-D tile, tensor_dim0_stride = NP1, and TDM
//         pad_interval/pad_amount inject 1 dword of padding per 32-dword row
//         => LDS stride 33 floats, bank (33l+i)%64 conflict-free, zero VALU
//         address math.  Tracked by TENSORcnt (in-order per wave).
//       - STORE: TDM cannot de-pad on store (ISA 8.2), and an unpadded LDS
//         layout would make ds_store a 32-way bank conflict; so stores use
//         32x global_store_async_from_lds_b32 (contiguous 128 B per instr),
//         tracked by ASYNCcnt with pointer-increment addressing.
//   * 3-buffer ring per wave: TDM-load(t+1) and async-stores(t-1) both
//     overlap compute(t).  s_wait_asynccnt 0x20 retires only the older store
//     batch (<=32 newer ones outstanding); s_wait_tensorcnt 1 admits the
//     prefetch while guaranteeing tile t is resident.
//   * Waves fully independent: no barriers, only counter waits.
// ---------------------------------------------------------------------------

#define TK             32                    // time steps per tile (32 dwords)
#define ROWS_PER_WAVE  32                    // lane == row within wave
#define WAVES_PER_BLK  4
#define BLOCK          (WAVES_PER_BLK * 32)  // 128 threads
#define LDS_STRIDE     33                    // TK + 1 pad dword (TDM-inserted)
#define SLICE          (ROWS_PER_WAVE * LDS_STRIDE)
#define NBUF           3

__device__ __forceinline__ uint32_t lds_offset(const float* p) {
    // Generic pointers to LDS carry the workgroup-relative LDS byte offset
    // in their low 32 bits on gfx1250.
    return (uint32_t)(uintptr_t)p;
}

// ---- Tensor Data Mover: one 2-D tile (32 rows x TK f32) -> padded LDS -----
__device__ __forceinline__ void tdm_load_tile(uint32_t lds_byte,
                                              const float* gsrc, int NP1) {
    typedef uint32_t v4u __attribute__((ext_vector_type(4)));
    typedef uint32_t v8u __attribute__((ext_vector_type(8)));
    const uint64_t ga = (uint64_t)(uintptr_t)gsrc;

    v4u g0;
    g0[0] = 1u;                                          // count=1, user D#
    g0[1] = lds_byte;                                    // lds_addr (bytes)
    g0[2] = (uint32_t)ga;                                // global_addr[31:0]
    g0[3] = (uint32_t)((ga >> 32) & 0x01FFFFFFu)         // global_addr[56:32]
          | 0x80000000u;                                 // type=2 ("image")

    v8u g1;
    g1[0] = (2u << 16)        // data_size = 2 -> 4-byte elements
          | (1u << 20)        // pad_enable
          | (4u << 22)        // pad_interval = 4 -> every 32 dwords
          | (0u << 25);       // pad_amount   = 0 -> 1 dword
    g1[1] = (uint32_t)TK << 16;             // tensor_dim0[15:0]   (=TK)
    g1[2] = (uint32_t)ROWS_PER_WAVE << 16;  // tensor_dim1[15:0]   (=32)
    g1[3] = (uint32_t)TK << 16;             // tile_dim0           (=TK)
    g1[4] = (uint32_t)ROWS_PER_WAVE;        // tile_dim1 (=32), tile_dim2=0
    g1[5] = (uint32_t)NP1;                  // tensor_dim0_stride lo32 (elems)
    g1[6] = 0u;                             // stride hi / dim1_stride lo
    g1[7] = 0u;                             // dim1_stride hi (unused, 2-D)

    asm volatile("tensor_load_to_lds %0, %1" :: "s"(g0), "s"(g1) : "memory");
}

__device__ __forceinline__ void async_store_b32(float* gdst, uint32_t lds_byte) {
    asm volatile("global_store_async_from_lds_b32 %0, %1, off"
                 :: "v"(gdst), "v"(lds_byte) : "memory");
}

__device__ __forceinline__ void wait_async_le32() {
    asm volatile("s_wait_asynccnt 0x20" ::: "memory");   // <=32 outstanding
}
__device__ __forceinline__ void wait_async0() {
    asm volatile("s_wait_asynccnt 0x0" ::: "memory");
}
__device__ __forceinline__ void wait_ds0() {
    asm volatile("s_wait_dscnt 0x0" ::: "memory");
}

__global__ __launch_bounds__(BLOCK)
void em_scan_kernel(const float* __restrict__ Z0,
                    const float* __restrict__ W,
                    const float* __restrict__ Wf,
                    const float* __restrict__ Wg,
                    float* __restrict__ Z,
                    int N) {
    __shared__ float tile[NBUF][WAVES_PER_BLK][SLICE];   // ~50 KB / block

    const int lane    = threadIdx.x & 31;
    const int wave    = threadIdx.x >> 5;
    const int row     = blockIdx.x * BLOCK + threadIdx.x;
    const int rowBase = blockIdx.x * BLOCK + wave * 32;
    const int NP1     = N + 1;

    const float dt  = 1.0f / (float)N;                   // T0=0, T1=1
    const float sdt = sqrtf(dt);
    const float c0  = 1.0f + Wf[0] * dt;                 // uniform scalar loads
    const float c1  = Wg[0] * sdt;

    const int numT = N / TK;                             // 1024/32 = 32 tiles

    // Wave-uniform global base of the wave's row block (col 1 of first row).
    const float* wtile0 = W + (size_t)rowBase * NP1 + 1;

    // ---- prologue: TDM-load tile 0 into ring buffer 0 ----------------------
    tdm_load_tile(lds_offset(&tile[0][wave][0]), wtile0, NP1);

    // initial condition Z[:,0] = Z0 (one 4 B store per thread, once)
    float z = Z0[row];
    Z[(size_t)row * NP1] = z;

    for (int t = 0; t < numT; ++t) {
        const int buf = t % NBUF;

        // 1) retire the *older* store batch (tile t-2) so its buffer can be
        //    refilled; <=32 newer stores (tile t-1) may remain in flight.
        wait_async_le32();

        // 2) prefetch tile t+1 via TDM (overlaps compute of tile t), then
        // 3) admit at most that one outstanding tensor op: tile t is resident.
        if (t + 1 < numT) {
            tdm_load_tile(lds_offset(&tile[(t + 1) % NBUF][wave][0]),
                          wtile0 + (size_t)(t + 1) * TK, NP1);
            __builtin_amdgcn_s_wait_tensorcnt(1);
        } else {
            __builtin_amdgcn_s_wait_tensorcnt(0);
        }

        // 4) compute: serial scan over TK steps, in place in padded LDS
        float* slice = &tile[buf][wave][lane * LDS_STRIDE];  // this lane's row
        #pragma unroll
        for (int i = 0; i < TK; ++i) {
            float w = slice[i];                      // ds_load_b32, conflict-free
            z = z * fmaf(c1, w, c0);                 // z *= (c0 + c1*w)
            slice[i] = z;                            // ds_store_b32
        }

        // 5) LDS writes visible to the async store engine
        wait_ds0();

        // 6) async stores: contiguous 128 B per instruction, pointer-stepped
        {
            float*   gz   = Z + (size_t)rowBase * NP1 + 1 + (size_t)t * TK + lane;
            uint32_t ldsb = lds_offset(&tile[buf][wave][lane]);
            for (int r = 0; r < ROWS_PER_WAVE; ++r) {
                async_store_b32(gz, ldsb);
                gz   += NP1;                         // next row, same column
                ldsb += LDS_STRIDE * 4;
            }
        }
    }

    wait_async0();   // drain stores (s_endpgm also implies wait-idle)
}

extern "C" void kernel_launch(void* const* d_in, const int* in_sizes, int n_in,
                              void* d_out, int out_size, void* d_ws, size_t ws_size,
                              hipStream_t stream) {
    const float* Z0 = (const float*)d_in[0];
    const float* W  = (const float*)d_in[1];
    const float* Wf = (const float*)d_in[2];
    const float* Wg = (const float*)d_in[3];
    float*       Z  = (float*)d_out;

    const int B   = in_sizes[0];            // 131072 paths
    const int NP1 = in_sizes[1] / B;        // 1025
    const int N   = NP1 - 1;                // 1024

    dim3 grid(B / BLOCK), block(BLOCK);
    em_scan_kernel<<<grid, block, 0, stream>>>(Z0, W, Wf, Wg, Z, N);

    // Reference returns the tuple (Z, W): second output is W passed through.
    const size_t elems = (size_t)B * (size_t)NP1;
    if ((size_t)out_size >= 2 * elems) {
        hipMemcpyAsync((float*)d_out + elems, W, elems * sizeof(float),
                       hipMemcpyDeviceToDevice, stream);
    }
}